// DecoderBlock_19894288515514
// MI455X (gfx1250) — compile-verified
//
#include <hip/hip_runtime.h>
#include <math.h>

// ---------------- CDNA5 WMMA types ----------------
typedef __attribute__((ext_vector_type(2))) float v2f;
typedef __attribute__((ext_vector_type(8))) float v8f;

__device__ __forceinline__ v8f wmma_f32(v2f a, v2f b, v8f c) {
    // D = A(16x4 f32) * B(4x16 f32) + C(16x16 f32)
    return __builtin_amdgcn_wmma_f32_16x16x4_f32(false, a, false, b, (short)0, c, false, false);
}

// Async global->LDS DMA (CDNA5, tracked by ASYNCcnt). GVS mode:
// mem_addr = SGPR base + 32-bit VGPR offset; LDS byte addr per-lane in first VGPR.
__device__ __forceinline__ void async_copy_b128(unsigned lds_byte, unsigned glb_byte_off,
                                                const float* base) {
    asm volatile("global_load_async_to_lds_b128 %0, %1, %2"
                 :: "v"(lds_byte), "v"(glb_byte_off), "s"(base)
                 : "memory");
}

// ---------------- constants ----------------
#define BQ 2
#define SEQ 2048
#define NST 512
#define NHEAD 8
#define DEPTH 64
#define ROWS (BQ * SEQ)          // 4096
#define QKV_STRIDE (3 * NST)     // 1536
#define MAXSEQ 2048

// LDS tile strides (floats); 16B-aligned rows for B128 async writes,
// odd-ish mod-64 spread to limit bank conflicts.
#define AS_STR 36
#define BS_STR 68

// =================================================================
// LayerNorm: one block (128 threads) per row of 512
// =================================================================
__global__ __launch_bounds__(128) void ln_kernel(const float* __restrict__ x,
                                                 const float* __restrict__ g,
                                                 const float* __restrict__ be,
                                                 float* __restrict__ y) {
    __shared__ float red1[4], red2[4];
    const int row = blockIdx.x;
    const int tid = threadIdx.x;
    const int lane = tid & 31, wid = tid >> 5;
    const float* xr = x + (size_t)row * NST;
    float xv[4], s1 = 0.f, s2 = 0.f;
#pragma unroll
    for (int i = 0; i < 4; ++i) {
        xv[i] = xr[tid + i * 128];
        s1 += xv[i];
        s2 += xv[i] * xv[i];
    }
#pragma unroll
    for (int off = 16; off > 0; off >>= 1) {
        s1 += __shfl_xor(s1, off);
        s2 += __shfl_xor(s2, off);
    }
    if (lane == 0) { red1[wid] = s1; red2[wid] = s2; }
    __syncthreads();
    s1 = red1[0] + red1[1] + red1[2] + red1[3];
    s2 = red2[0] + red2[1] + red2[2] + red2[3];
    const float mu = s1 * (1.f / NST);
    const float var = s2 * (1.f / NST) - mu * mu;
    const float rstd = rsqrtf(var + 1e-5f);
    float* yr = y + (size_t)row * NST;
#pragma unroll
    for (int i = 0; i < 4; ++i) {
        const int c = tid + i * 128;
        yr[c] = (xv[i] - mu) * rstd * g[c] + be[c];
    }
}

// =================================================================
// f32 WMMA GEMM: out[M,N] = A[M,K] @ W[K,N] + bias  (+epilogue)
// 64x64 tile per workgroup (4 waves, each 32x32 = 2x2 WMMA tiles).
// Double-buffered async global->LDS pipeline: next K-tile DMA overlaps
// the current tile's WMMAs; wave blocks only on s_wait_asynccnt 8.
// EPI: 0 = bias, 1 = bias + residual add, 2 = bias + tanh-GELU
// =================================================================
template <int EPI>
__global__ __launch_bounds__(128) void gemm_f32_wmma(const float* __restrict__ A,
                                                     const float* __restrict__ W,
                                                     const float* __restrict__ bias,
                                                     const float* __restrict__ res,
                                                     float* __restrict__ out,
                                                     int M, int N, int K) {
    __shared__ float As[2][64][AS_STR];
    __shared__ float Bs[2][32][BS_STR];
    const int tid = threadIdx.x;
    const int lane = tid & 31, wid = tid >> 5;
    const int lo = lane & 15, hi = lane >> 4;
    const int wm = wid >> 1, wn = wid & 1;
    const int m0 = blockIdx.x * 64, n0 = blockIdx.y * 64;

    const unsigned as_lds = (unsigned)(size_t)(&As[0][0][0]);
    const unsigned bs_lds = (unsigned)(size_t)(&Bs[0][0][0]);

    // Issue one 64x32 A-tile + 32x64 B-tile as 8 B128 async DMAs per wave.
    auto issue_tile = [&](int p, int k0) {
#pragma unroll
        for (int i = 0; i < 4; ++i) {              // A: 512 chunks of 4 floats
            const int q = tid + i * 128;
            const int r = q >> 3, c4 = (q & 7) * 4;
            const unsigned lds = as_lds + (unsigned)(((p * 64 + r) * AS_STR + c4) * 4);
            const unsigned off = (unsigned)(((m0 + r) * K + k0 + c4) * 4);
            async_copy_b128(lds, off, A);
        }
#pragma unroll
        for (int i = 0; i < 4; ++i) {              // B: 512 chunks of 4 floats
            const int q = tid + i * 128;
            const int r = q >> 4, c4 = (q & 15) * 4;
            const unsigned lds = bs_lds + (unsigned)(((p * 32 + r) * BS_STR + c4) * 4);
            const unsigned off = (unsigned)(((k0 + r) * N + n0 + c4) * 4);
            async_copy_b128(lds, off, W);
        }
    };

    v8f acc[2][2] = {};
    const int nk = K / 32;
    issue_tile(0, 0);

    for (int kt = 0; kt < nk; ++kt) {
        const int p = kt & 1;
        if (kt + 1 < nk) {
            issue_tile(p ^ 1, (kt + 1) * 32);
            asm volatile("s_wait_asynccnt 0x8" ::: "memory");   // current tile done, next in flight
        } else {
            asm volatile("s_wait_asynccnt 0x0" ::: "memory");
        }
        __syncthreads();
#pragma unroll
        for (int kk = 0; kk < 32; kk += 4) {
            const int ka = kk + 2 * hi;
            v2f af[2], bf[2];
#pragma unroll
            for (int mt = 0; mt < 2; ++mt) {
                const int rr = wm * 32 + mt * 16 + lo;
                af[mt].x = As[p][rr][ka];
                af[mt].y = As[p][rr][ka + 1];
            }
#pragma unroll
            for (int nt = 0; nt < 2; ++nt) {
                const int cc = wn * 32 + nt * 16 + lo;
                bf[nt].x = Bs[p][ka][cc];
                bf[nt].y = Bs[p][ka + 1][cc];
            }
#pragma unroll
            for (int mt = 0; mt < 2; ++mt)
#pragma unroll
                for (int nt = 0; nt < 2; ++nt)
                    acc[mt][nt] = wmma_f32(af[mt], bf[nt], acc[mt][nt]);
        }
        __syncthreads();
    }
    // epilogue: C layout lane L vgpr r -> row r+8*hi, col L%16
#pragma unroll
    for (int mt = 0; mt < 2; ++mt)
#pragma unroll
        for (int nt = 0; nt < 2; ++nt)
#pragma unroll
            for (int r = 0; r < 8; ++r) {
                const int m = m0 + wm * 32 + mt * 16 + r + 8 * hi;
                const int n = n0 + wn * 32 + nt * 16 + lo;
                float v = acc[mt][nt][r] + bias[n];
                if (EPI == 1) v += res[(size_t)m * N + n];
                if (EPI == 2) {
                    const float u = 0.7978845608028654f * (v + 0.044715f * v * v * v);
                    v = 0.5f * v * (1.f + tanhf(u));
                }
                out[(size_t)m * N + n] = v;
            }
}

// =================================================================
// Flash attention with relative-position skew.
// One wave per workgroup handles one (b, h, 16-query) tile.
// logit[q,k] = ( q.k + q.E[s-1-q+k] ) / sqrt(D), causal mask.
// =================================================================
__global__ __launch_bounds__(32) void attn_kernel(const float* __restrict__ qkv,
                                                  const float* __restrict__ E,
                                                  float* __restrict__ att) {
    __shared__ float ps[16 * 16];
    const int lane = threadIdx.x & 31;
    const int lo = lane & 15, hi = lane >> 4;
    const int wg = blockIdx.x;
    const int qt = wg & 127;          // S/16 tiles
    const int h = (wg >> 7) & 7;
    const int b = wg >> 10;
    const int q0 = qt * 16;

    const float* qbase = qkv + (size_t)b * SEQ * QKV_STRIDE;
    const float* Eh = E + ((size_t)h * MAXSEQ + (MAXSEQ - SEQ)) * DEPTH;

    // preload Q fragments: A layout lane L: M=L%16, K pair 2*hi(+4t)
    v2f qa[16];
    {
        const float* qrow = qbase + (size_t)(q0 + lo) * QKV_STRIDE + h * DEPTH;
#pragma unroll
        for (int t = 0; t < 16; ++t)
            qa[t] = *(const v2f*)(qrow + 4 * t + 2 * hi);
    }

    v8f acc_o[4] = {};
    float mrow[8], lrow[8];
#pragma unroll
    for (int r = 0; r < 8; ++r) { mrow[r] = -3.0e38f; lrow[r] = 0.f; }

    for (int k0 = 0; k0 <= q0; k0 += 16) {
        v8f s = {}, r0a = {}, r1a = {};
        const int lb = SEQ - 16 - q0 + k0;               // >= 0
        const float* krow = qbase + (size_t)(k0 + lo) * QKV_STRIDE + NST + h * DEPTH;
        const float* e0row = Eh + (size_t)(lb + lo) * DEPTH;
        const int l1 = lb + 16 + lo;
        const float* e1row = Eh + (size_t)l1 * DEPTH;
        const bool e1ok = (l1 < SEQ);
        // prefetch next K/V tile rows into cache (global_prefetch_b8)
        if (k0 + 16 <= q0) {
            __builtin_prefetch(krow + (size_t)16 * QKV_STRIDE, 0, 3);
            __builtin_prefetch(krow + (size_t)16 * QKV_STRIDE + NST, 0, 3);
        }
#pragma unroll
        for (int t = 0; t < 16; ++t) {
            const int d = 4 * t + 2 * hi;
            const v2f kb = *(const v2f*)(krow + d);
            const v2f e0 = *(const v2f*)(e0row + d);
            v2f e1;
            if (e1ok) e1 = *(const v2f*)(e1row + d);
            else { e1.x = 0.f; e1.y = 0.f; }
            s = wmma_f32(qa[t], kb, s);
            r0a = wmma_f32(qa[t], e0, r0a);
            r1a = wmma_f32(qa[t], e1, r1a);
        }

        // gather skewed rel via ds_bpermute, scale, causal mask
        float p[8];
#pragma unroll
        for (int r = 0; r < 8; ++r) {
            const int i = r + 8 * hi;
            const int c = 15 - i + lo;                       // 0..30
            const int idx = (((lane & 16) | (c & 15)) << 2); // byte index
            const float v0 = __int_as_float(__builtin_amdgcn_ds_bpermute(idx, __float_as_int(r0a[r])));
            const float v1 = __int_as_float(__builtin_amdgcn_ds_bpermute(idx, __float_as_int(r1a[r])));
            const float rel = (c < 16) ? v0 : v1;
            float sc = (s[r] + rel) * 0.125f;                // 1/sqrt(64)
            if (k0 + lo > q0 + i) sc = -1.0e9f;              // causal mask
            p[r] = sc;
        }

        // online softmax update
#pragma unroll
        for (int r = 0; r < 8; ++r) {
            float mx = p[r];
            mx = fmaxf(mx, __shfl_xor(mx, 1));
            mx = fmaxf(mx, __shfl_xor(mx, 2));
            mx = fmaxf(mx, __shfl_xor(mx, 4));
            mx = fmaxf(mx, __shfl_xor(mx, 8));
            const float mnew = fmaxf(mrow[r], mx);
            const float scale = __expf(mrow[r] - mnew);
            const float pe = __expf(p[r] - mnew);
            float sum = pe;
            sum += __shfl_xor(sum, 1);
            sum += __shfl_xor(sum, 2);
            sum += __shfl_xor(sum, 4);
            sum += __shfl_xor(sum, 8);
            lrow[r] = lrow[r] * scale + sum;
            mrow[r] = mnew;
#pragma unroll
            for (int ct = 0; ct < 4; ++ct) acc_o[ct][r] *= scale;
            ps[(r + 8 * hi) * 16 + lo] = pe;                 // stage P for transpose
        }
        __syncthreads();

        // O += P @ V   (A = P 16x16 via LDS, B = V 16x64 from global)
        const float* vbase = qbase + 2 * NST + h * DEPTH;
#pragma unroll
        for (int t2 = 0; t2 < 4; ++t2) {
            const int kk = 4 * t2 + 2 * hi;
            const v2f pa = *(const v2f*)(&ps[lo * 16 + kk]);
#pragma unroll
            for (int ct = 0; ct < 4; ++ct) {
                v2f vb;
                vb.x = vbase[(size_t)(k0 + kk) * QKV_STRIDE + ct * 16 + lo];
                vb.y = vbase[(size_t)(k0 + kk + 1) * QKV_STRIDE + ct * 16 + lo];
                acc_o[ct] = wmma_f32(pa, vb, acc_o[ct]);
            }
        }
        __syncthreads();
    }

    // normalize and store att in (B, S, N_STATE) layout
    float* obase = att + ((size_t)(b * SEQ + q0)) * NST + h * DEPTH;
#pragma unroll
    for (int r = 0; r < 8; ++r) {
        const float inv = 1.f / lrow[r];
        const int i = r + 8 * hi;
#pragma unroll
        for (int ct = 0; ct < 4; ++ct)
            obase[(size_t)i * NST + ct * 16 + lo] = acc_o[ct][r] * inv;
    }
}

// =================================================================
// present[b, {k,v}, h, s, d] <- qkv
// =================================================================
__global__ __launch_bounds__(256) void present_kernel(const float* __restrict__ qkv,
                                                      float* __restrict__ present) {
    const int idx = blockIdx.x * 256 + threadIdx.x;   // 2^22 total
    const int d = idx & 63;
    const int sidx = (idx >> 6) & 2047;
    const int h = (idx >> 17) & 7;
    const int t = (idx >> 20) & 1;
    const int b = idx >> 21;
    present[idx] = qkv[((size_t)(b * SEQ + sidx)) * QKV_STRIDE + (t + 1) * NST + h * DEPTH + d];
}

// =================================================================
// Host launch
// =================================================================
extern "C" void kernel_launch(void* const* d_in, const int* in_sizes, int n_in,
                              void* d_out, int out_size, void* d_ws, size_t ws_size,
                              hipStream_t stream) {
    (void)in_sizes; (void)n_in; (void)out_size; (void)ws_size;
    const float* x      = (const float*)d_in[0];
    // d_in[1] = mask (causal, applied analytically)
    const float* W_qkv  = (const float*)d_in[2];
    const float* b_qkv  = (const float*)d_in[3];
    const float* W_proj = (const float*)d_in[4];
    const float* b_proj = (const float*)d_in[5];
    const float* E      = (const float*)d_in[6];
    const float* g1     = (const float*)d_in[7];
    const float* be1    = (const float*)d_in[8];
    const float* g2     = (const float*)d_in[9];
    const float* be2    = (const float*)d_in[10];
    const float* W_fc1  = (const float*)d_in[11];
    const float* b_fc1  = (const float*)d_in[12];
    const float* W_fc2  = (const float*)d_in[13];
    const float* b_fc2  = (const float*)d_in[14];

    float* out     = (float*)d_out;
    float* xout    = out;                          // (B,S,N) -> also holds x_mid
    float* present = out + (size_t)ROWS * NST;     // (B,2,H,S,D)

    float* ws    = (float*)d_ws;
    float* lnbuf = ws;                                       // 4096*512
    float* qkv   = lnbuf + (size_t)ROWS * NST;               // 4096*1536
    float* attb  = qkv + (size_t)ROWS * QKV_STRIDE;          // 4096*512
    float* h1    = attb + (size_t)ROWS * NST;                // 4096*2048

    // a = LN1(x)
    ln_kernel<<<ROWS, 128, 0, stream>>>(x, g1, be1, lnbuf);
    // qkv = a @ W_qkv + b_qkv
    gemm_f32_wmma<0><<<dim3(ROWS / 64, (3 * NST) / 64), 128, 0, stream>>>(
        lnbuf, W_qkv, b_qkv, nullptr, qkv, ROWS, 3 * NST, NST);
    // present = stack(k, v)
    present_kernel<<<(BQ * 2 * NHEAD * SEQ * DEPTH) / 256, 256, 0, stream>>>(qkv, present);
    // attention (flash, fused relative skew + causal softmax)
    attn_kernel<<<BQ * NHEAD * (SEQ / 16), 32, 0, stream>>>(qkv, E, attb);
    // x_mid = x + att @ W_proj + b_proj   (stored in d_out x region)
    gemm_f32_wmma<1><<<dim3(ROWS / 64, NST / 64), 128, 0, stream>>>(
        attb, W_proj, b_proj, x, xout, ROWS, NST, NST);
    // m = LN2(x_mid)
    ln_kernel<<<ROWS, 128, 0, stream>>>(xout, g2, be2, lnbuf);
    // h1 = gelu(m @ W_fc1 + b_fc1)
    gemm_f32_wmma<2><<<dim3(ROWS / 64, (4 * NST) / 64), 128, 0, stream>>>(
        lnbuf, W_fc1, b_fc1, nullptr, h1, ROWS, 4 * NST, NST);
    // x_out = x_mid + h1 @ W_fc2 + b_fc2
    gemm_f32_wmma<1><<<dim3(ROWS / 64, NST / 64), 128, 0, stream>>>(
        h1, W_fc2, b_fc2, xout, xout, ROWS, NST, 4 * NST);
}